// MutualInformationEstimator_70677981823030
// MI455X (gfx1250) — compile-verified
//
#include <hip/hip_runtime.h>
#include <math.h>
#include <stdint.h>

#define ZD     18      // joint dims
#define TILE   512     // rows per LDS tile
#define RS     20      // padded row stride in floats (80B, 16B-aligned rows)
#define BLOCK  512     // threads per block = 16 waves
#define WPB    16      // queries (waves) per block
#define K_NEIGH 11
#define KP1    12      // k+1 smallest kept

__device__ __forceinline__ double dev_digamma(double x) {
  double r = 0.0;
  while (x < 6.0) { r -= 1.0 / x; x += 1.0; }
  double f = 1.0 / (x * x);
  double t = f * (-1.0/12.0 + f * (1.0/120.0 + f * (-1.0/252.0 +
             f * (1.0/240.0 + f * (-1.0/132.0)))));
  return log(x) - 0.5 / x + t + r;
}

__device__ __forceinline__ float rfl_f32(float v) {
  union { float f; int i; } u; u.f = v;
  u.i = __builtin_amdgcn_readfirstlane(u.i);
  return u.f;
}

__device__ __forceinline__ float max3(float a, float b, float c) {
  return fmaxf(fmaxf(a, b), c);
}

// ---------------- scale factors: s_d = 1/(std_d + 1e-10); mean cancels -------
__global__ void mi_scales_kernel(const float* __restrict__ actual,
                                 const float* __restrict__ predicted,
                                 float* __restrict__ scales, int n) {
  __shared__ double ssum[256];
  __shared__ double ssq[256];
  const int col = blockIdx.x;                 // 0..17
  const float* base = (col < 9) ? actual : predicted;
  const int c = (col < 9) ? col : col - 9;
  double s = 0.0, q = 0.0;
  for (int i = threadIdx.x; i < n; i += 256) {
    double v = (double)base[i * 9 + c];
    s += v; q += v * v;
  }
  ssum[threadIdx.x] = s; ssq[threadIdx.x] = q;
  __syncthreads();
  for (int off = 128; off > 0; off >>= 1) {
    if ((int)threadIdx.x < off) {
      ssum[threadIdx.x] += ssum[threadIdx.x + off];
      ssq[threadIdx.x]  += ssq[threadIdx.x + off];
    }
    __syncthreads();
  }
  if (threadIdx.x == 0) {
    double mean = ssum[0] / (double)n;
    double var  = ssq[0] / (double)n - mean * mean;
    if (var < 0.0) var = 0.0;
    scales[col] = (float)(1.0 / (sqrt(var) + 1e-10));
  }
}

// ---------------- main: one wave per query; async-to-LDS tile streaming ------
__global__ void __launch_bounds__(BLOCK)
mi_main_kernel(const float* __restrict__ actual,
               const float* __restrict__ predicted,
               const float* __restrict__ scales,
               double* __restrict__ terms, int n) {
  __shared__ __align__(16) float tile[2][TILE * RS];

  const int tid  = threadIdx.x;
  const int wave = tid >> 5;
  const int lane = tid & 31;
  const int q    = blockIdx.x * WPB + wave;
  const int nTiles = n / TILE;            // 32
  const int totalIters = 2 * nTiles;      // pass1 (eps) + pass2 (counts)

  // scales & scaled query row -> SGPRs (uniform per wave)
  float ns[ZD], qs[ZD];
  #pragma unroll
  for (int d = 0; d < ZD; ++d) {
    float sv = rfl_f32(scales[d]);
    float zq = (d < 9) ? actual[q * 9 + d] : predicted[q * 9 + (d - 9)];
    ns[d] = -sv;                       // term = |fma(-s, z_j, s*z_q)|
    qs[d] = rfl_f32(sv * zq);
  }

  // Hoisted per-thread staging offsets (loop-invariant across all 64 tiles).
  // Element e = tid + m*BLOCK enumerates (row,col) with e = row*9+col, so the
  // global byte offset is tileBase*36 + e*4 (no division in the hot path).
  unsigned ldsOff[9], gOff[9];
  #pragma unroll
  for (int m = 0; m < 9; ++m) {
    int e = tid + m * BLOCK;
    int row = e / 9, col = e - row * 9;
    ldsOff[m] = (unsigned)(row * RS + col) * 4u;
    gOff[m]   = (unsigned)e * 4u;
  }

  auto issue_tile = [&](int tileIdx, int buf) {
    const unsigned ldsBase = (unsigned)(uintptr_t)(&tile[buf][0]);
    const unsigned gbase   = (unsigned)(tileIdx * TILE * 9) * 4u;
    #pragma unroll
    for (int m = 0; m < 9; ++m) {                       // 'actual' half
      unsigned lds = ldsBase + ldsOff[m];
      unsigned go  = gbase + gOff[m];
      asm volatile("global_load_async_to_lds_b32 %0, %1, %2"
                   :: "v"(lds), "v"(go), "s"(actual) : "memory");
    }
    #pragma unroll
    for (int m = 0; m < 9; ++m) {                       // 'predicted' half
      unsigned lds = ldsBase + ldsOff[m] + 36u;         // cols 9..17
      unsigned go  = gbase + gOff[m];
      asm volatile("global_load_async_to_lds_b32 %0, %1, %2"
                   :: "v"(lds), "v"(go), "s"(predicted) : "memory");
    }
  };

  float a[KP1];                      // sorted ascending, 12 smallest so far
  #pragma unroll
  for (int m = 0; m < KP1; ++m) a[m] = 3.4e38f;
  int cnt[6] = {0, 0, 0, 0, 0, 0};
  float eps = 0.0f;

  issue_tile(0, 0);                  // pipeline prologue

  for (int t = 0; t < totalIters; ++t) {
    const int buf = t & 1;
    if (t + 1 < totalIters) {
      int ti = t + 1;
      if (ti >= nTiles) ti -= nTiles;                   // avoid runtime modulo
      issue_tile(ti, (t + 1) & 1);
      asm volatile("s_wait_asynccnt 18" ::: "memory");  // tile t complete
    } else {
      asm volatile("s_wait_asynccnt 0" ::: "memory");
    }
    __syncthreads();                 // everyone's copies visible

    if (t == nTiles) {
      // merge per-lane 12-smallest across the wave -> eps = 12th smallest
      #pragma unroll
      for (int rsel = 0; rsel < KP1; ++rsel) {
        float h = a[0];
        #pragma unroll
        for (int off = 16; off > 0; off >>= 1)
          h = fminf(h, __shfl_xor(h, off, 32));
        unsigned long long bm = __ballot(a[0] == h);
        int src = __ffsll(bm) - 1;
        if (lane == src) {           // consume one instance
          #pragma unroll
          for (int m = 0; m < KP1 - 1; ++m) a[m] = a[m + 1];
          a[KP1 - 1] = 3.4e38f;
        }
        eps = h;
      }
    }

    const float* tp = &tile[buf][0];
    if (t < nTiles) {
      // ---- pass 1: joint-space Chebyshev, keep 12 smallest ----
      #pragma unroll 2
      for (int i = 0; i < TILE / 32; ++i) {
        const float* rp = tp + (lane + i * 32) * RS;
        float4 A  = *(const float4*)(rp);
        float4 B  = *(const float4*)(rp + 4);
        float4 C  = *(const float4*)(rp + 8);
        float4 Dv = *(const float4*)(rp + 12);
        float2 E  = *(const float2*)(rp + 16);
        float zr[ZD] = {A.x,A.y,A.z,A.w, B.x,B.y,B.z,B.w, C.x,C.y,C.z,C.w,
                        Dv.x,Dv.y,Dv.z,Dv.w, E.x,E.y};
        float u[ZD];
        #pragma unroll
        for (int d = 0; d < ZD; ++d)
          u[d] = fabsf(fmaf(ns[d], zr[d], qs[d]));
        // balanced max tree (depth ~5 instead of an 18-deep chain)
        float m0 = fmaxf(u[0], u[1]),  m1 = fmaxf(u[2], u[3]);
        float m2 = fmaxf(u[4], u[5]),  m3 = fmaxf(u[6], u[7]);
        float m4 = fmaxf(u[8], u[9]),  m5 = fmaxf(u[10], u[11]);
        float m6 = fmaxf(u[12], u[13]), m7 = fmaxf(u[14], u[15]);
        float m8 = fmaxf(u[16], u[17]);
        float n0 = fmaxf(m0, m1), n1 = fmaxf(m2, m3);
        float n2 = fmaxf(m4, m5), n3 = fmaxf(m6, m7);
        float dmax = fmaxf(fmaxf(fmaxf(n0, n1), fmaxf(n2, n3)), m8);
        if (dmax < a[KP1 - 1]) {     // wave-coherent skip when nobody inserts
          float x = dmax;
          #pragma unroll
          for (int m = 0; m < KP1; ++m) {
            float lo = fminf(a[m], x);
            x = fmaxf(a[m], x);
            a[m] = lo;
          }
        }
      }
    } else {
      // ---- pass 2: 6 marginal-group counts within eps ----
      #pragma unroll 2
      for (int i = 0; i < TILE / 32; ++i) {
        const float* rp = tp + (lane + i * 32) * RS;
        float4 A  = *(const float4*)(rp);
        float4 B  = *(const float4*)(rp + 4);
        float4 C  = *(const float4*)(rp + 8);
        float4 Dv = *(const float4*)(rp + 12);
        float2 E  = *(const float2*)(rp + 16);
        float zr[ZD] = {A.x,A.y,A.z,A.w, B.x,B.y,B.z,B.w, C.x,C.y,C.z,C.w,
                        Dv.x,Dv.y,Dv.z,Dv.w, E.x,E.y};
        #pragma unroll
        for (int g = 0; g < 6; ++g) {
          float u0 = fmaf(ns[3*g+0], zr[3*g+0], qs[3*g+0]);
          float u1 = fmaf(ns[3*g+1], zr[3*g+1], qs[3*g+1]);
          float u2 = fmaf(ns[3*g+2], zr[3*g+2], qs[3*g+2]);
          float gm = max3(fabsf(u0), fabsf(u1), fabsf(u2));
          cnt[g] += (gm <= eps) ? 1 : 0;
        }
      }
    }
    __syncthreads();                 // tile consumed before its buffer reloads
  }

  // wave-reduce counts; lane 0 writes the per-query digamma sum (deterministic)
  #pragma unroll
  for (int g = 0; g < 6; ++g) {
    int c = cnt[g];
    #pragma unroll
    for (int off = 16; off > 0; off >>= 1) c += __shfl_xor(c, off, 32);
    cnt[g] = c;
  }
  if (lane == 0) {
    double term = 0.0;
    #pragma unroll
    for (int g = 0; g < 6; ++g) {
      int c = cnt[g] - 1;            // remove self
      if (c < 1) c = 1;
      term += dev_digamma((double)c + 1.0);
    }
    terms[q] = term;
  }
}

// ---------------- finalize: fixed-order deterministic reduction --------------
__global__ void mi_finalize_kernel(const double* __restrict__ terms,
                                   float* __restrict__ out, int n) {
  __shared__ double sred[256];
  double s = 0.0;
  for (int i = threadIdx.x; i < n; i += 256) s += terms[i];
  sred[threadIdx.x] = s;
  __syncthreads();
  for (int off = 128; off > 0; off >>= 1) {
    if ((int)threadIdx.x < off) sred[threadIdx.x] += sred[threadIdx.x + off];
    __syncthreads();
  }
  if (threadIdx.x == 0) {
    double mi = dev_digamma((double)n) + dev_digamma((double)K_NEIGH)
              - sred[0] / (double)n;
    out[0] = (float)(mi > 0.0 ? mi : 0.0);
  }
}

extern "C" void kernel_launch(void* const* d_in, const int* in_sizes, int n_in,
                              void* d_out, int out_size, void* d_ws, size_t ws_size,
                              hipStream_t stream) {
  const float* actual    = (const float*)d_in[0];
  const float* predicted = (const float*)d_in[1];
  const int n = in_sizes[0] / 9;           // 16384

  float*  scales = (float*)d_ws;                       // 18 floats
  double* terms  = (double*)((char*)d_ws + 128);       // n doubles

  mi_scales_kernel<<<ZD, 256, 0, stream>>>(actual, predicted, scales, n);
  mi_main_kernel<<<n / WPB, BLOCK, 0, stream>>>(actual, predicted, scales, terms, n);
  mi_finalize_kernel<<<1, 256, 0, stream>>>(terms, (float*)d_out, n);
}